// SparsePatternA2_42752104465039
// MI455X (gfx1250) — compile-verified
//
#include <hip/hip_runtime.h>
#include <hip/hip_bf16.h>
#include <math.h>

#define BB    2
#define CC    256
#define HH    64
#define NN    4096
#define KTOP  1228
#define NREST (NN - KTOP)
#define MIPC  16
#define QD    16   // q=0..14 Taylor terms, q=15 = "rest" channel

typedef __attribute__((ext_vector_type(2))) float v2f;
typedef __attribute__((ext_vector_type(8))) float v8f;

__constant__ float c_invfact[16] = {
  1.0f, 1.0f, 5.0e-1f, 1.66666666666667e-1f, 4.16666666666667e-2f,
  8.33333333333333e-3f, 1.38888888888889e-3f, 1.98412698412698e-4f,
  2.48015873015873e-5f, 2.75573192239859e-6f, 2.75573192239859e-7f,
  2.50521083854417e-8f, 2.08767569878681e-9f, 1.60590438368216e-10f,
  1.14707455977297e-11f, 0.0f };

// ---- k1: row means, col means, per-(b,c) total sum -------------------------
__global__ void k1_means(const float* __restrict__ x, float* __restrict__ rm,
                         float* __restrict__ cm, float* __restrict__ sumall) {
  int bc = blockIdx.x;                       // b*C + c
  const float* xp = x + (size_t)bc * NN;
  __shared__ float rs[HH], cs[HH];
  int t = threadIdx.x;                       // 256 threads
  if (t < HH) rs[t] = 0.f;
  else if (t < 2 * HH) cs[t - HH] = 0.f;
  __syncthreads();
  int h = t >> 2, off = (t & 3) * 16;
  const float* row = xp + h * HH + off;
  float rpart = 0.f;
  #pragma unroll
  for (int j = 0; j < 16; ++j) {
    float v = row[j];
    rpart += v;
    atomicAdd(&cs[off + j], v);
  }
  atomicAdd(&rs[h], rpart);
  __syncthreads();
  if (t < HH) rm[(size_t)bc * HH + t] = rs[t] * (1.f / HH);
  else if (t < 2 * HH) cm[(size_t)bc * HH + (t - HH)] = cs[t - HH] * (1.f / HH);
  if (t == 0) {
    float s = 0.f;
    for (int i = 0; i < HH; ++i) s += rs[i];
    sumall[bc] = s;
  }
}

// ---- k2: conv1x1 + BN + ReLU + heads -> importance (rank-1) ----------------
__global__ void k2_importance(const float* __restrict__ rm, const float* __restrict__ cm,
    const float* __restrict__ w1, const float* __restrict__ b1,
    const float* __restrict__ g,  const float* __restrict__ be,
    const float* __restrict__ mu, const float* __restrict__ var,
    const float* __restrict__ wh, const float* __restrict__ bh,
    const float* __restrict__ ww, const float* __restrict__ bw,
    const float* __restrict__ wd1, const float* __restrict__ bd1,
    const float* __restrict__ wd2, const float* __restrict__ bd2,
    float* __restrict__ imp) {
  int b = blockIdx.x;
  int t = threadIdx.x;                       // 256 threads; l = t
  __shared__ float w1L[MIPC * CC];
  __shared__ float y2L[MIPC * 256];
  __shared__ float aH[64], aW[64], aD1[64], aD2[64];
  for (int i = t; i < MIPC * CC; i += 256) w1L[i] = w1[i];
  __syncthreads();
  int l = t, seg = l >> 6, pos = l & 63;
  // segments: 0->rowmean, 1->colmean, 2,3->rowmean (rot90 branches collapse)
  const float* src = (seg == 1) ? (cm + (size_t)(b * CC) * HH)
                                : (rm + (size_t)(b * CC) * HH);
  float acc[MIPC];
  #pragma unroll
  for (int m = 0; m < MIPC; ++m) acc[m] = 0.f;
  for (int c = 0; c < CC; ++c) {
    float v = src[c * HH + pos];
    #pragma unroll
    for (int m = 0; m < MIPC; ++m) acc[m] += w1L[m * CC + c] * v;
  }
  #pragma unroll
  for (int m = 0; m < MIPC; ++m) {
    float sc = g[m] * rsqrtf(var[m] + 1e-5f);
    float y  = (acc[m] + b1[m] - mu[m]) * sc + be[m];
    y2L[m * 256 + l] = fmaxf(y, 0.f);
  }
  __syncthreads();
  {
    int head = t >> 6, p = t & 63;
    const float* w = (head == 0) ? wh : (head == 1) ? ww : (head == 2) ? wd1 : wd2;
    float bias = (head == 0) ? bh[0] : (head == 1) ? bw[0] : (head == 2) ? bd1[0] : bd2[0];
    float s = bias;
    #pragma unroll
    for (int m = 0; m < MIPC; ++m) s += w[m] * y2L[m * 256 + head * 64 + p];
    float a = 1.f / (1.f + expf(-s));
    if (head == 0) aH[p] = a; else if (head == 1) aW[p] = a;
    else if (head == 2) aD1[p] = a; else aD2[p] = a;
  }
  __syncthreads();
  for (int n = t; n < NN; n += 256) {
    int h = n >> 6, w = n & 63;
    imp[(size_t)b * NN + n] = aH[h] * (aW[w] * aD1[w] * aD2[w]);
  }
}

// ---- k3a: exact stable ranking -> top-k membership -------------------------
__global__ void k3a_rank(const float* __restrict__ imp, float* __restrict__ impm,
                         float* __restrict__ mask0) {
  int b = blockIdx.y;
  int j = blockIdx.x * 512 + threadIdx.x;
  __shared__ float impL[NN];
  for (int i = threadIdx.x; i < NN; i += 512) impL[i] = imp[(size_t)b * NN + i];
  __syncthreads();
  float v = impL[j];
  int rank = 0;
  for (int m = 0; m < NN; ++m) {
    float u = impL[m];
    rank += (int)((u > v) || ((u == v) && (m < j)));   // lax.top_k tie rule
  }
  bool in = rank < KTOP;
  impm[(size_t)b * NN + j]  = in ? v : 0.f;
  mask0[(size_t)b * NN + j] = in ? 1.f : 0.f;
}

// ---- k3b: power sums S_q and per-row inverse softmax denominator -----------
__global__ void k3b_coef(const float* __restrict__ imp, const float* __restrict__ impm,
                         const float* __restrict__ maskp, float* __restrict__ Sws,
                         float* __restrict__ invden) {
  int b = blockIdx.x;
  int t = threadIdx.x;                       // 256 threads
  __shared__ float imA[NN];                  // masked imp
  __shared__ float imB[NN];                  // raw imp
  __shared__ float S[16];
  for (int i = t; i < NN; i += 256) { imA[i] = impm[(size_t)b * NN + i];
                                      imB[i] = imp [(size_t)b * NN + i]; }
  if (t < 16) S[t] = 0.f;
  __syncthreads();
  float part[15];
  #pragma unroll
  for (int q = 0; q < 15; ++q) part[q] = 0.f;
  for (int n = t; n < NN; n += 256) {
    float u = imA[n], pw = u;
    #pragma unroll
    for (int q = 1; q < 15; ++q) { part[q] += pw; pw *= u; }
  }
  #pragma unroll
  for (int q = 1; q < 15; ++q) atomicAdd(&S[q], part[q]);
  __syncthreads();
  if (t == 0) { S[0] = (float)KTOP; S[15] = (float)NREST; }
  __syncthreads();
  if (t < 16) Sws[b * 16 + t] = S[t];
  float sgm = 1.f / (1.f + expf(-maskp[0]));
  for (int i = t; i < NN; i += 256) {
    float ci = sgm * imB[i];
    float den = (float)NREST, pw = 1.f;
    #pragma unroll
    for (int q = 0; q < 15; ++q) { den += pw * c_invfact[q] * S[q]; pw *= ci; }
    invden[(size_t)b * NN + i] = 1.f / den;
  }
}

// ---- k4: moment matrix M[c,q] = sum_{j in T} imp_j^q x[c,j] ----------------
__global__ void k4_M(const float* __restrict__ x, const float* __restrict__ impm,
                     const float* __restrict__ mask0, const float* __restrict__ sumall,
                     float* __restrict__ Mws) {
  int b = blockIdx.y;
  int c0 = blockIdx.x * 64;
  int t = threadIdx.x;                       // 256 threads
  __shared__ float xt[64 * 129];             // padded (bank spread)
  __shared__ float imL[128], m0L[128];
  __shared__ float Mred[64][16];
  if (t < 64) {
    #pragma unroll
    for (int q = 0; q < 16; ++q) Mred[t][q] = 0.f;
  }
  float acc[15];
  #pragma unroll
  for (int q = 0; q < 15; ++q) acc[q] = 0.f;
  int cl = t >> 2, sub = t & 3;
  for (int nc = 0; nc < NN; nc += 128) {
    __syncthreads();
    if (t < 128) { imL[t] = impm[(size_t)b * NN + nc + t];
                   m0L[t] = mask0[(size_t)b * NN + nc + t]; }
    for (int idx = t; idx < 64 * 128; idx += 256) {
      int r = idx >> 7, ccx = idx & 127;
      xt[r * 129 + ccx] = x[((size_t)(b * CC + c0 + r)) * NN + nc + ccx];
    }
    __syncthreads();
    for (int ccx = sub; ccx < 128; ccx += 4) {
      float xv = xt[cl * 129 + ccx];
      float u = imL[ccx];
      acc[0] += m0L[ccx] * xv;
      float pw = u;
      #pragma unroll
      for (int q = 1; q < 15; ++q) { acc[q] += pw * xv; pw *= u; }
    }
  }
  __syncthreads();
  #pragma unroll
  for (int q = 0; q < 15; ++q) atomicAdd(&Mred[cl][q], acc[q]);
  __syncthreads();
  if (t < 64) {
    int c = c0 + t;
    float* Mp = Mws + ((size_t)(b * CC + c)) * QD;
    #pragma unroll
    for (int q = 0; q < 15; ++q) Mp[q] = Mred[t][q];
    Mp[15] = sumall[b * CC + c] - Mred[t][0];  // "rest" channel
  }
}

// ---- k5: WMMA f32 reconstruction: out[c,i] = (sum_q A[i,q] M[c,q]) / D_i ---
__global__ void k5_out(const float* __restrict__ imp, const float* __restrict__ invden,
                       const float* __restrict__ Mws, const float* __restrict__ maskp,
                       float* __restrict__ out) {
  int b = blockIdx.y;
  int iblk = blockIdx.x * 128;
  int t = threadIdx.x;                       // 256 threads = 8 wave32
  __shared__ float Ml[CC * 17];              // M[c,q], stride 17 (bank spread)
  __shared__ float cf[128 * 17];             // coef A[i,q]
  __shared__ float ivL[128];
  float sgm = 1.f / (1.f + expf(-maskp[0]));
  for (int idx = t; idx < CC * QD; idx += 256) {
    int c = idx >> 4, q = idx & 15;
    Ml[c * 17 + q] = Mws[(size_t)(b * CC + c) * QD + q];
  }
  if (t < 128) ivL[t] = invden[(size_t)b * NN + iblk + t];
  {
    int il = t & 127, rep = t >> 7;
    float ci = sgm * imp[(size_t)b * NN + iblk + il];
    int q0 = rep * 8;
    float pw = 1.f;
    for (int q = 0; q < q0; ++q) pw *= ci;
    #pragma unroll
    for (int qq = 0; qq < 8; ++qq) {
      int q = q0 + qq;
      cf[il * 17 + q] = (q == 15) ? 1.f : pw * c_invfact[q];
      pw *= ci;
    }
  }
  __syncthreads();
  int wv = t >> 5, lane = t & 31;
  int li = lane & 15, hi = lane >> 4;
  int iL = wv * 16 + li;                     // local column index (n dim)
  float ivd = ivL[iL];
  // B matrix (K=q rows, N=i cols); layout mirrors A: VGPR0={K0|K2}, VGPR1={K1|K3}
  v2f bop[4];
  #pragma unroll
  for (int tt = 0; tt < 4; ++tt) {
    int kq = tt * 4 + (hi ? 2 : 0);
    bop[tt].x = cf[iL * 17 + kq];
    bop[tt].y = cf[iL * 17 + kq + 1];
  }
  for (int ct = 0; ct < CC / 16; ++ct) {
    int c = ct * 16 + li;                    // A-matrix row (M dim = channel)
    v8f acc = {};
    #pragma unroll
    for (int tt = 0; tt < 4; ++tt) {
      int kq = tt * 4 + (hi ? 2 : 0);
      v2f a;
      a.x = Ml[c * 17 + kq];
      a.y = Ml[c * 17 + kq + 1];
      acc = __builtin_amdgcn_wmma_f32_16x16x4_f32(false, a, false, bop[tt],
                                                  (short)0, acc, false, false);
    }
    // D layout: lane holds col n=li, rows r + 8*hi -> contiguous stores in n
    #pragma unroll
    for (int r = 0; r < 8; ++r) {
      int cr = ct * 16 + r + hi * 8;
      out[(size_t)(b * CC + cr) * NN + iblk + wv * 16 + li] = acc[r] * ivd;
    }
  }
}

// ---- host-side launcher ----------------------------------------------------
extern "C" void kernel_launch(void* const* d_in, const int* in_sizes, int n_in,
                              void* d_out, int out_size, void* d_ws, size_t ws_size,
                              hipStream_t stream) {
  const float* x    = (const float*)d_in[0];
  const float* w1   = (const float*)d_in[1];
  const float* b1   = (const float*)d_in[2];
  const float* g    = (const float*)d_in[3];
  const float* be   = (const float*)d_in[4];
  const float* mu   = (const float*)d_in[5];
  const float* var  = (const float*)d_in[6];
  const float* wh   = (const float*)d_in[7];
  const float* bh   = (const float*)d_in[8];
  const float* ww   = (const float*)d_in[9];
  const float* bw   = (const float*)d_in[10];
  const float* wd1  = (const float*)d_in[11];
  const float* bd1  = (const float*)d_in[12];
  const float* wd2  = (const float*)d_in[13];
  const float* bd2  = (const float*)d_in[14];
  const float* mask = (const float*)d_in[15];

  float* ws      = (float*)d_ws;
  float* rm      = ws;                 // B*C*64 = 32768
  float* cm      = ws + 32768;         // 32768
  float* sumall  = ws + 65536;         // 512
  float* imp     = ws + 66048;         // 8192
  float* impm    = ws + 74240;         // 8192
  float* mask0   = ws + 82432;         // 8192
  float* Sws     = ws + 90624;         // 32
  float* invden  = ws + 90656;         // 8192
  float* Mws     = ws + 98848;         // 8192
  (void)in_sizes; (void)n_in; (void)out_size; (void)ws_size;

  k1_means<<<BB * CC, 256, 0, stream>>>(x, rm, cm, sumall);
  k2_importance<<<BB, 256, 0, stream>>>(rm, cm, w1, b1, g, be, mu, var,
                                        wh, bh, ww, bw, wd1, bd1, wd2, bd2, imp);
  dim3 g3a(NN / 512, BB);
  k3a_rank<<<g3a, 512, 0, stream>>>(imp, impm, mask0);
  k3b_coef<<<BB, 256, 0, stream>>>(imp, impm, mask, Sws, invden);
  dim3 g4(CC / 64, BB);
  k4_M<<<g4, 256, 0, stream>>>(x, impm, mask0, sumall, Mws);
  dim3 g5(NN / 128, BB);
  k5_out<<<g5, 256, 0, stream>>>(imp, invden, Mws, mask, (float*)d_out);
}